// MessageAggregator_6468220748479
// MI455X (gfx1250) — compile-verified
//
#include <hip/hip_runtime.h>

#define E_N   32768
#define NODES 2048
#define D     128
#define F_DIM 32
#define DCAT  160

typedef float v2f __attribute__((ext_vector_type(2)));
typedef float v8f __attribute__((ext_vector_type(8)));

__device__ __forceinline__ float log_sig(float x) {
    // numerically stable log(sigmoid(x)) = min(x,0) - log1p(exp(-|x|))
    return fminf(x, 0.f) - log1pf(expf(-fabsf(x)));
}

// s_wait_asynccnt 0 : wait for all outstanding async global->LDS transfers
__device__ __forceinline__ void wait_async0() {
#if defined(__has_builtin) && __has_builtin(__builtin_amdgcn_s_wait_asynccnt)
    __builtin_amdgcn_s_wait_asynccnt(0);
#else
    asm volatile("s_wait_asynccnt 0x0" ::: "memory");
#endif
}

// LDS byte offset of a workgroup-shared object: flat LDS-aperture addresses carry
// the allocation-relative offset in addr[31:0] (same convention as async VDST).
__device__ __forceinline__ unsigned lds_off_of(const void* p) {
    return (unsigned)(unsigned long long)p;
}

// Async global->LDS, 8 bytes per lane, LDS destination fully scattered per lane.
template<bool NT>
__device__ __forceinline__ void async_ld_b64(unsigned lds_off, unsigned long long gaddr) {
    if (NT)
        asm volatile("global_load_async_to_lds_b64 %0, %1, off th:TH_LOAD_NT"
                     :: "v"(lds_off), "v"(gaddr) : "memory");
    else
        asm volatile("global_load_async_to_lds_b64 %0, %1, off"
                     :: "v"(lds_off), "v"(gaddr) : "memory");
}
__device__ __forceinline__ void async_ld_b32(unsigned lds_off, unsigned long long gaddr) {
    asm volatile("global_load_async_to_lds_b32 %0, %1, off"
                 :: "v"(lds_off), "v"(gaddr) : "memory");
}

// Generic fp32 WMMA GEMM: out[M x 128] = op(A[M x K] @ B)
//   BT=1 : B is a weight matrix stored row-major [128 x K]; effective B[k][n] = W[n*ldb + k]
//   BT=0 : B is row-major [K x 128]
//   BIAS : add bias[n];  ACT : log_sigmoid;  SUB : subtract sub[m*128+n] first
//   NTA  : non-temporal streaming hint for A (the 256MB mask operands)
// Block = 256 threads = 8 wave32s; block computes a 128x128 tile, wave w owns rows
// [w*16,w*16+16) x all 8 col tiles. K streamed through LDS in 64-wide zero-padded
// chunks (handles K=160). Chunks are filled with GLOBAL_LOAD_ASYNC_TO_LDS (ASYNCcnt),
// writing the K-pair-interleaved float2 swizzle directly, so every WMMA fragment is a
// single aligned ds_load_b64 straight into the operand VGPR pair:
//   A: row stride 34 float2 (68 dw = 4 mod 64)        -> conflict-free b64 frag reads
//   B: pair-row stride 144 float2 (288 dw = 32 mod 64) -> conflict-free b64 frag reads
template<bool BT, bool BIAS, bool ACT, bool SUB, bool NTA>
__global__ __launch_bounds__(256)
void gemm_wmma_f32(const float* __restrict__ A, int lda,
                   const float* __restrict__ B, int ldb,
                   const float* __restrict__ bias,
                   const float* __restrict__ sub,
                   float* __restrict__ out, int ldo, int ocol0,
                   int M, int K)
{
    __shared__ v2f Als2[128 * 34];   // 128 rows x 32 K-pairs (+2 pad)  ~34.8 KB
    __shared__ v2f Bls2[32 * 144];   // 32 K-pairs x 128 N (+16 pad)    ~36.9 KB

    const int tid   = threadIdx.x;
    const int lane  = tid & 31;
    const int wave  = tid >> 5;
    const int m0blk = blockIdx.x * 128;
    const int wm0   = wave * 16;
    const int lr    = lane & 15;   // M index (A frag) / N index (B frag)
    const int lh    = lane >> 4;   // half-wave selects K pair

    v8f acc[8] = {};

    for (int kc = 0; kc < K; kc += 64) {
        // A chunk 128x64 -> 128x32 float2, async straight into swizzled LDS
        #pragma unroll
        for (int i = 0; i < 16; ++i) {
            int flat = i * 256 + tid;          // 0..4095
            int r = flat >> 5, c2 = flat & 31; // row, K-pair
            int gk = kc + 2 * c2;
            unsigned loff = lds_off_of(&Als2[r * 34 + c2]);
            if (gk < K) {                      // K even -> pair fully in range
                async_ld_b64<NTA>(loff,
                    (unsigned long long)(A + (size_t)(m0blk + r) * lda + gk));
            } else {
                Als2[r * 34 + c2] = v2f{0.f, 0.f};   // zero pad (ds_store)
            }
        }
        // B chunk 64x128 -> 32 K-pair rows of 128 float2
        #pragma unroll
        for (int i = 0; i < 16; ++i) {
            int flat = i * 256 + tid;
            int pr = flat >> 7, n = flat & 127; // K-pair row, col
            int gk = kc + 2 * pr;
            unsigned loff = lds_off_of(&Bls2[pr * 144 + n]);
            if (gk < K) {
                if (BT) {
                    async_ld_b64<false>(loff,
                        (unsigned long long)(B + (size_t)n * ldb + gk));
                } else {
                    async_ld_b32(loff,
                        (unsigned long long)(B + (size_t)gk * ldb + n));
                    async_ld_b32(loff + 4,
                        (unsigned long long)(B + (size_t)(gk + 1) * ldb + n));
                }
            } else {
                Bls2[pr * 144 + n] = v2f{0.f, 0.f};
            }
        }
        wait_async0();        // all async LDS writes by this wave landed
        __syncthreads();      // + DScnt for zero-pad stores,全wave visibility

        #pragma unroll 4
        for (int kk = 0; kk < 64; kk += 4) {
            const int kp = (kk >> 1) + lh;                 // this lane's K pair
            v2f a = Als2[(wm0 + lr) * 34 + kp];            // one ds_load_b64
            #pragma unroll
            for (int nt = 0; nt < 8; ++nt) {
                v2f b = Bls2[kp * 144 + nt * 16 + lr];     // one ds_load_b64
                acc[nt] = __builtin_amdgcn_wmma_f32_16x16x4_f32(
                    false, a, false, b, (short)0, acc[nt], false, false);
            }
        }
        __syncthreads();
    }

    // Epilogue. C layout: VGPR v, lanes0-15 -> M=v, lanes16-31 -> M=v+8; N = lane%16
    #pragma unroll
    for (int nt = 0; nt < 8; ++nt) {
        #pragma unroll
        for (int v = 0; v < 8; ++v) {
            float x = acc[nt][v];
            int gm = m0blk + wm0 + v + 8 * lh;
            int gn = nt * 16 + lr;
            if (SUB)  x -= sub[(size_t)gm * 128 + gn];
            if (BIAS) x += bias[gn];
            if (ACT)  x = log_sig(x);
            out[(size_t)gm * ldo + ocol0 + gn] = x;
        }
    }
}

// cat[:, 128:160] = feature
__global__ void copy_feature(const float* __restrict__ f, float* __restrict__ cat) {
    int idx = blockIdx.x * blockDim.x + threadIdx.x;   // 0 .. E*F-1
    int e = idx >> 5;       // F_DIM == 32
    int c = idx & 31;
    cat[(size_t)e * DCAT + 128 + c] = f[idx];
}

extern "C" void kernel_launch(void* const* d_in, const int* in_sizes, int n_in,
                              void* d_out, int out_size, void* d_ws, size_t ws_size,
                              hipStream_t stream)
{
    (void)in_sizes; (void)n_in; (void)out_size; (void)ws_size;
    const float* state = (const float*)d_in[0];
    const float* feat  = (const float*)d_in[1];
    const float* mask  = (const float*)d_in[2];
    const float* maskT = (const float*)d_in[3];
    const float* W1m   = (const float*)d_in[4];
    const float* b1m   = (const float*)d_in[5];
    const float* W2m   = (const float*)d_in[6];
    const float* W1a   = (const float*)d_in[7];
    const float* b1a   = (const float*)d_in[8];
    const float* W2a   = (const float*)d_in[9];
    float* out = (float*)d_out;

    // workspace layout (fp32): h[E*128] | s[E*128] | nsum[NODES*128] | cat[E*160]
    float* h    = (float*)d_ws;
    float* s    = h    + (size_t)E_N * D;
    float* nsum = s    + (size_t)E_N * D;
    float* cat  = nsum + (size_t)NODES * D;

    dim3 blk(256);

    // 1) h = ls(state @ W1m^T + b1m)
    gemm_wmma_f32<true, true, true, false, false><<<E_N / 128, blk, 0, stream>>>(
        state, D, W1m, D, b1m, nullptr, h, D, 0, E_N, D);
    // 2) s = ls(h @ W2m^T)
    gemm_wmma_f32<true, false, true, false, false><<<E_N / 128, blk, 0, stream>>>(
        h, D, W2m, D, nullptr, nullptr, s, D, 0, E_N, D);
    // 3) nsum = mask @ s   (mask streamed non-temporally, 256MB read once)
    gemm_wmma_f32<false, false, false, false, true><<<NODES / 128, blk, 0, stream>>>(
        mask, E_N, s, D, nullptr, nullptr, nsum, D, 0, NODES, E_N);
    // 4) cat[:,128:160] = feature
    copy_feature<<<(E_N * F_DIM) / 256, blk, 0, stream>>>(feat, cat);
    // 5) cat[:,0:128] = maskT @ nsum - s   (maskT streamed non-temporally)
    gemm_wmma_f32<false, false, false, true, true><<<E_N / 128, blk, 0, stream>>>(
        maskT, NODES, nsum, D, nullptr, s, cat, DCAT, 0, E_N, NODES);
    // 6) h = ls(cat @ W1a^T + b1a)   (K=160, zero-padded chunks)
    gemm_wmma_f32<true, true, true, false, false><<<E_N / 128, blk, 0, stream>>>(
        cat, DCAT, W1a, DCAT, b1a, nullptr, h, D, 0, E_N, DCAT);
    // 7) out = ls(h @ W2a^T)
    gemm_wmma_f32<true, false, true, false, false><<<E_N / 128, blk, 0, stream>>>(
        h, D, W2a, D, nullptr, nullptr, out, D, 0, E_N, D);
}